// ModelKGCL_6949257085589
// MI455X (gfx1250) — compile-verified
//
#include <hip/hip_runtime.h>
#include <stdint.h>

// -------- problem constants --------
#define DDIM 64
#define MNBR 40
#define NEG_INF_F (-9000000000000000.0f)
#define LEAK 0.2f

// -------- tiling --------
#define TS    68     // LDS row stride (floats): 272B rows -> 16B aligned, banks (4m+d)%64
#define ROWS  48     // 40 entity rows + item row at 40 + pad rows 41..47 for WMMA group 2
#define WAVES 4
#define THREADS (WAVES * 32)

typedef float v2f __attribute__((ext_vector_type(2)));
typedef float v8f __attribute__((ext_vector_type(8)));
typedef int   v4i __attribute__((ext_vector_type(4)));

typedef __attribute__((address_space(1))) v4i g_v4i;   // global int4
typedef __attribute__((address_space(3))) v4i l_v4i;   // LDS int4

__device__ __forceinline__ g_v4i* glob_b128(const void* p) {
  return (g_v4i*)(uintptr_t)p;
}
__device__ __forceinline__ l_v4i* lds_b128(const void* p) {
  // low 32 bits of a generic shared pointer == LDS byte offset (ISA 10.2)
  return (l_v4i*)(uint32_t)(uintptr_t)p;
}

#if __has_builtin(__builtin_amdgcn_wmma_f32_16x16x4_f32)
#define USE_WMMA 1
#else
#define USE_WMMA 0
#endif

#if __has_builtin(__builtin_amdgcn_global_load_async_to_lds_b128)
#define USE_ASYNC 1
#else
#define USE_ASYNC 0
#endif

#if __has_builtin(__builtin_amdgcn_s_wait_asynccnt)
#define ASYNC_WAIT() __builtin_amdgcn_s_wait_asynccnt(0)
#else
#define ASYNC_WAIT() asm volatile("s_wait_asynccnt 0" ::: "memory")
#endif

// Precompute w1 = W @ a[:64], w2 = W @ a[64:]   (two 64-vectors)
__global__ void kgat_prep(const float* __restrict__ W, const float* __restrict__ a,
                          float* __restrict__ w12) {
  int t = threadIdx.x;                 // 0..127
  int row = t & 63;
  const float* av = a + ((t >> 6) ? DDIM : 0);
  float s = 0.f;
  for (int e = 0; e < DDIM; ++e) s += W[row * DDIM + e] * av[e];
  w12[t] = s;                          // [0..63]=w1, [64..127]=w2
}

__global__ void __launch_bounds__(THREADS)
kgat_main(const float* __restrict__ item, const float* __restrict__ ent,
          const int* __restrict__ adj, const float* __restrict__ w12,
          float* __restrict__ out, int N) {
  __shared__ float w1s[DDIM];
  __shared__ float w2s[DDIM];
  __shared__ float tile[WAVES][ROWS * TS];
  __shared__ float eb[WAVES][64];

  const int tid  = threadIdx.x;
  const int lane = tid & 31;
  const int wv   = tid >> 5;

  if (tid < DDIM)            w1s[tid] = w12[tid];
  else if (tid < 2 * DDIM)   w2s[tid - DDIM] = w12[tid];
  __syncthreads();

  const int h   = lane >> 4;   // wave half (selects K offset in WMMA fragments)
  const int l16 = lane & 15;

  // per-lane register caches (constant across items)
  const float w1r0 = w1s[lane], w1r1 = w1s[lane + 32];
#if USE_WMMA
  float w2r[32];
#pragma unroll
  for (int i = 0; i < 16; ++i) {       // B fragment values for K-step i: K = 4i + 2h, +1
    w2r[2 * i]     = w2s[4 * i + 2 * h];
    w2r[2 * i + 1] = w2s[4 * i + 2 * h + 1];
  }
#endif

  float* mytile = tile[wv];
  float* myeb   = eb[wv];

  const int nwaves = gridDim.x * WAVES;
  for (int n = blockIdx.x * WAVES + wv; n < N; n += nwaves) {
    const float* eg = ent + (size_t)n * (MNBR * DDIM);

    // ---- stage E (40x64) + item row into LDS, padded rows (read once from HBM) ----
#if USE_ASYNC
#pragma unroll
    for (int i = 0; i < 20; ++i) {
      int chunk = lane + 32 * i;             // 640 b128 chunks
      int r = chunk >> 4, c4 = (chunk & 15) << 2;
      __builtin_amdgcn_global_load_async_to_lds_b128(
          glob_b128(eg + r * DDIM + c4),
          lds_b128(mytile + r * TS + c4), 0, 0);
    }
    if (lane < 16) {
      __builtin_amdgcn_global_load_async_to_lds_b128(
          glob_b128(item + (size_t)n * DDIM + 4 * lane),
          lds_b128(mytile + MNBR * TS + 4 * lane), 0, 0);
    }
    ASYNC_WAIT();
#else
#pragma unroll
    for (int i = 0; i < 20; ++i) {
      int chunk = lane + 32 * i;
      int r = chunk >> 4, c4 = (chunk & 15) << 2;
      const float4 v = *(const float4*)(eg + r * DDIM + c4);
      *(float4*)(mytile + r * TS + c4) = v;
    }
    if (lane < 16) {
      const float4 v = *(const float4*)(item + (size_t)n * DDIM + 4 * lane);
      *(float4*)(mytile + MNBR * TS + 4 * lane) = v;
    }
#endif
    __builtin_amdgcn_wave_barrier();

    // ---- e0 = item . w1 (cross-lane reduce) ----
    float p = mytile[MNBR * TS + lane] * w1r0 + mytile[MNBR * TS + 32 + lane] * w1r1;
#pragma unroll
    for (int s = 16; s; s >>= 1) p += __shfl_xor(p, s, 32);
    const float e0 = p;

    // ---- e[m] = E[m,:] . w2  (exact f32) ----
#if USE_WMMA
    // Three 16-row groups; B columns all equal w2-slice, so D[i,*] = e[16g+i].
#pragma unroll
    for (int g = 0; g < 3; ++g) {
      v8f c = {0.f, 0.f, 0.f, 0.f, 0.f, 0.f, 0.f, 0.f};
      const float* arow = mytile + (16 * g + l16) * TS + 2 * h;  // A: K = v + 2h
#pragma unroll
      for (int kb = 0; kb < 16; ++kb) {
        v2f av, bv;
        av.x = arow[4 * kb];
        av.y = arow[4 * kb + 1];
        bv.x = w2r[2 * kb];
        bv.y = w2r[2 * kb + 1];
        c = __builtin_amdgcn_wmma_f32_16x16x4_f32(false, av, false, bv,
                                                  (short)0, c, false, false);
      }
      if (l16 == 0) {                      // D: M = v + 8h, value uniform over N
#pragma unroll
        for (int v = 0; v < 8; ++v) myeb[16 * g + 8 * h + v] = c[v];
      }
    }
#else
    {
      float acc = 0.f;
#pragma unroll 8
      for (int d = 0; d < DDIM; ++d) acc += mytile[lane * TS + d] * w2s[d];
      myeb[lane] = acc;
      if (lane < 8) {
        float acc2 = 0.f;
#pragma unroll 8
        for (int d = 0; d < DDIM; ++d) acc2 += mytile[(32 + lane) * TS + d] * w2s[d];
        myeb[32 + lane] = acc2;
      }
    }
#endif
    __builtin_amdgcn_wave_barrier();

    // ---- leaky-relu, adjacency mask, stable softmax over m=0..39 ----
    float ev1 = myeb[lane] + e0;
    ev1 = ev1 > 0.f ? ev1 : LEAK * ev1;
    ev1 = (adj[(size_t)n * MNBR + lane] > 0) ? ev1 : NEG_INF_F;
    float ev2 = NEG_INF_F;
    if (lane < 8) {
      ev2 = myeb[32 + lane] + e0;
      ev2 = ev2 > 0.f ? ev2 : LEAK * ev2;
      ev2 = (adj[(size_t)n * MNBR + 32 + lane] > 0) ? ev2 : NEG_INF_F;
    }
    float mx = fmaxf(ev1, ev2);
#pragma unroll
    for (int s = 16; s; s >>= 1) mx = fmaxf(mx, __shfl_xor(mx, s, 32));
    float p1 = __expf(ev1 - mx);
    float p2 = (lane < 8) ? __expf(ev2 - mx) : 0.f;
    float den = p1 + p2;
#pragma unroll
    for (int s = 16; s; s >>= 1) den += __shfl_xor(den, s, 32);
    const float inv = 1.f / den;
    myeb[lane] = p1 * inv;
    if (lane < 8) myeb[32 + lane] = p2 * inv;
    __builtin_amdgcn_wave_barrier();

    // ---- agg = att @ E ; out = agg + item (bank-conflict-free reads) ----
    float agg0 = 0.f, agg1 = 0.f;
#pragma unroll 8
    for (int m = 0; m < MNBR; ++m) {
      const float am = myeb[m];
      agg0 += am * mytile[m * TS + lane];
      agg1 += am * mytile[m * TS + 32 + lane];
    }
    out[(size_t)n * DDIM + lane]      = agg0 + mytile[MNBR * TS + lane];
    out[(size_t)n * DDIM + 32 + lane] = agg1 + mytile[MNBR * TS + 32 + lane];
    __builtin_amdgcn_wave_barrier();   // protect tile/eb reuse next iteration
  }
}

extern "C" void kernel_launch(void* const* d_in, const int* in_sizes, int n_in,
                              void* d_out, int out_size, void* d_ws, size_t ws_size,
                              hipStream_t stream) {
  const float* item = (const float*)d_in[0];
  const float* ent  = (const float*)d_in[1];
  const int*   adj  = (const int*)d_in[2];
  const float* W    = (const float*)d_in[3];
  const float* a    = (const float*)d_in[4];
  float* out = (float*)d_out;
  float* w12 = (float*)d_ws;          // 128 floats of scratch
  const int N = in_sizes[0] / DDIM;

  hipLaunchKernelGGL(kgat_prep, dim3(1), dim3(128), 0, stream, W, a, w12);

  int blocks = 1024;                   // 4096 waves, ~7 items each; persistent loop
  int maxb = (N + WAVES - 1) / WAVES;
  if (blocks > maxb) blocks = maxb;
  hipLaunchKernelGGL(kgat_main, dim3(blocks), dim3(THREADS), 0, stream,
                     item, ent, adj, w12, out, N);
}